// Wavetable_1597727834408
// MI455X (gfx1250) — compile-verified
//
#include <hip/hip_runtime.h>
#include <math.h>

// Wavetable synth for MI455X (gfx1250, wave32).
//
// Pipeline (hierarchical scan, full-device parallel):
//   K1: per-(row,chunk) partial sums of f0                     -> ws.partials
//   K2: per-row exclusive scan of 32 chunk sums (wave scan)    -> ws.offsets
//   K3: per-chunk: WMMA-based segmented inclusive scan of f0,
//       add offsets, phase = frac(cumsum/SR), 2-tap wavetable
//       interpolation from LDS, vectorized b128 stores.
//
// The intra-wave scan of 256 f32 values is done as a true matrix product
// D = T(16x16 lower-triangular ones) * X(16x16) using 4 chained
// v_wmma_f32_16x16x4_f32 (f32 in, f32 out -> exact 1.0*x products, plain
// f32 accumulation, same precision class as the reference cumsum).

typedef __attribute__((ext_vector_type(2))) float v2f;
typedef __attribute__((ext_vector_type(8))) float v8f;

#define T_LEN   64000
#define NROWS   8
#define CHUNK   2048
#define NCHUNK  32          // ceil(64000/2048); last chunk holds 512 valid
#define WTL     512
#define INV_SR  (1.0f / 16000.0f)
#define KSCALE  (512.0f / 511.0f)

// ---------------- Kernel 1: per-chunk partial sums ----------------
__global__ void k_chunk_sums(const float* __restrict__ f0,
                             float* __restrict__ partials) {
    const int r = blockIdx.y, c = blockIdx.x;
    const int t = threadIdx.x;
    const int chunkValid = min(CHUNK, T_LEN - c * CHUNK);
    const float* row = f0 + (size_t)r * T_LEN + (size_t)c * CHUNK;

    float s = 0.0f;
    const int gi = t * 8;
    if (gi + 7 < chunkValid) {
        const float4* p = (const float4*)(row + gi);
        float4 a = p[0], b = p[1];
        s = (a.x + a.y) + (a.z + a.w) + (b.x + b.y) + (b.z + b.w);
    } else {
        #pragma unroll
        for (int j = 0; j < 8; ++j)
            if (gi + j < chunkValid) s += row[gi + j];
    }
    // wave32 reduction
    #pragma unroll
    for (int m = 16; m > 0; m >>= 1) s += __shfl_xor(s, m);

    __shared__ float lds[8];
    const int wv = t >> 5, lane = t & 31;
    if (lane == 0) lds[wv] = s;
    __syncthreads();
    if (t == 0) {
        float tot = 0.0f;
        #pragma unroll
        for (int w = 0; w < 8; ++w) tot += lds[w];
        partials[r * NCHUNK + c] = tot;
    }
}

// ---------------- Kernel 2: per-row exclusive scan of chunk sums ----------------
__global__ void k_scan_offsets(const float* __restrict__ partials,
                               float* __restrict__ offsets) {
    const int wv = threadIdx.x >> 5;      // one wave per row, 8 rows
    const int lane = threadIdx.x & 31;    // one lane per chunk, 32 chunks
    float v = partials[wv * NCHUNK + lane];
    float inc = v;
    #pragma unroll
    for (int s = 1; s < 32; s <<= 1) {
        float t = __shfl_up(inc, s);
        if (lane >= s) inc += t;
    }
    offsets[wv * NCHUNK + lane] = inc - v;   // exclusive
}

// ---------------- Kernel 3: WMMA scan + wavetable synthesis ----------------
__global__ void k_scan_synth(const float* __restrict__ f0,
                             const float* __restrict__ wavetable,
                             const float* __restrict__ offsets,
                             float* __restrict__ out) {
    __shared__ float xlds[CHUNK];   // chunk of f0, column-major 16x16 tiles
    __shared__ float wtlds[WTL];    // full wavetable
    __shared__ float wtot[8];       // per-wave totals

    const int r = blockIdx.y, c = blockIdx.x;
    const int t = threadIdx.x;
    const int chunkValid = min(CHUNK, T_LEN - c * CHUNK);
    const float* row = f0 + (size_t)r * T_LEN + (size_t)c * CHUNK;

    // stage chunk into LDS (zero-fill beyond row end so the scan is defined)
    const int gi = t * 8;
    if (gi + 7 < chunkValid) {
        const float4* p = (const float4*)(row + gi);
        ((float4*)xlds)[t * 2]     = p[0];
        ((float4*)xlds)[t * 2 + 1] = p[1];
    } else {
        #pragma unroll
        for (int j = 0; j < 8; ++j)
            xlds[gi + j] = (gi + j < chunkValid) ? row[gi + j] : 0.0f;
    }
    // stage wavetable
    wtlds[t] = wavetable[t];
    wtlds[t + 256] = wavetable[t + 256];
    __syncthreads();

    const int wv   = t >> 5;
    const int lane = t & 31;
    const int n    = lane & 15;     // column index (and A-matrix row M)
    const int hi   = lane >> 4;     // half-wave selector
    const float* xw = xlds + wv * 256;

    // D = T16 * X16 via 4 chained v_wmma_f32_16x16x4_f32.
    // A 16x4 f32 layout (ISA 7.12.2): lanes 0-15 -> K=0(V0),1(V1);
    //                                 lanes 16-31 -> K=2(V0),3(V1); M = lane&15.
    // B 4x16 assumed mirrored: low half K=0,1 / high half K=2,3; N = lane&15.
    // Column n of X = elements [n*16 .. n*16+15] of the wave's 256-elem segment.
    v8f acc = {};
    #pragma unroll
    for (int ck = 0; ck < 4; ++ck) {
        const int k0 = 4 * ck + 2 * hi;
        v2f a, b;
        a.x = (k0     <= n) ? 1.0f : 0.0f;   // T[m=n][k0]
        a.y = (k0 + 1 <= n) ? 1.0f : 0.0f;   // T[m=n][k0+1]
        b.x = xw[n * 16 + k0];
        b.y = xw[n * 16 + k0 + 1];
        acc = __builtin_amdgcn_wmma_f32_16x16x4_f32(
                  false, a, false, b, (short)0, acc, false, false);
    }
    // acc[rl] = D[(hi?8:0)+rl][n] = inclusive scan within column n.

    // cross-column fixup: column totals D[15][n] live at lane 16+n, VGPR7
    float colTot = __shfl(acc[7], 16 + n);
    float inc = colTot;
    #pragma unroll
    for (int s = 1; s < 16; s <<= 1) {
        float tv = __shfl(inc, (n >= s) ? (n - s) : n);
        if (n >= s) inc += tv;
    }
    const float colOff  = inc - colTot;
    const float waveTot = __shfl(inc, 15);

    // cross-wave + cross-chunk offsets
    if (lane == 0) wtot[wv] = waveTot;
    __syncthreads();
    float pre = offsets[r * NCHUNK + c] + colOff;
    for (int w = 0; w < wv; ++w) pre += wtot[w];

    // synthesize: phase = frac(cumsum/SR); 2-tap triangular interp.
    // wt_phase = linspace(0,1,512) -> grid step 1/511; weight_i =
    // relu(1 - |phase*511 - i| * 512/511); only floor(u), floor(u)+1 survive.
    float res[8];
    #pragma unroll
    for (int rl = 0; rl < 8; ++rl) {
        float cs = acc[rl] + pre;
        float tv = cs * INV_SR;
        float ph = tv - floorf(tv);          // [0,1) exactly in f32
        float u  = ph * 511.0f;              // < 511
        float fi = floorf(u);
        int   i0 = (int)fi;                  // <= 510
        float fr = u - fi;
        float w0 = fmaxf(0.0f, 1.0f - fr * KSCALE);
        float w1 = fmaxf(0.0f, 1.0f - (1.0f - fr) * KSCALE);
        res[rl] = wtlds[i0] * w0 + wtlds[i0 + 1] * w1;
    }

    // lane owns 8 contiguous outputs: element = n*16 + hi*8 + rl
    const int inChunk = wv * 256 + n * 16 + hi * 8;
    float* dst = out + (size_t)r * T_LEN + (size_t)c * CHUNK + inChunk;
    if (inChunk + 3 < chunkValid)
        *(float4*)dst = make_float4(res[0], res[1], res[2], res[3]);
    if (inChunk + 7 < chunkValid)
        *(float4*)(dst + 4) = make_float4(res[4], res[5], res[6], res[7]);
}

extern "C" void kernel_launch(void* const* d_in, const int* in_sizes, int n_in,
                              void* d_out, int out_size, void* d_ws, size_t ws_size,
                              hipStream_t stream) {
    (void)in_sizes; (void)n_in; (void)out_size; (void)ws_size;
    const float* f0        = (const float*)d_in[0];   // [8,1,64000] f32
    const float* wavetable = (const float*)d_in[1];   // [1,512,1,1] f32
    float* out      = (float*)d_out;                  // [8,1,64000] f32
    float* partials = (float*)d_ws;                   // 256 floats
    float* offsets  = (float*)d_ws + NROWS * NCHUNK;  // 256 floats

    dim3 grid(NCHUNK, NROWS);
    k_chunk_sums <<<grid, 256, 0, stream>>>(f0, partials);
    k_scan_offsets<<<1,    256, 0, stream>>>(partials, offsets);
    k_scan_synth <<<grid, 256, 0, stream>>>(f0, wavetable, offsets, out);
}